// EntropyEvaluator_87832081203327
// MI455X (gfx1250) — compile-verified
//
#include <hip/hip_runtime.h>
#include <hip/hip_bf16.h>

typedef float v2f __attribute__((ext_vector_type(2)));
typedef float v8f __attribute__((ext_vector_type(8)));

#define D_IN  512
#define D_H   128
#define D_OUT 64
#define NEG_SLOPE 0.01f

__device__ __forceinline__ float lrelu(float v) { return v >= 0.0f ? v : NEG_SLOPE * v; }

// ---------------- utility kernels ----------------

__global__ void fill_zero_u32(unsigned* __restrict__ p, int n) {
    int i = blockIdx.x * blockDim.x + threadIdx.x;
    if (i < n) p[i] = 0u;
}

__global__ void deg_count_kernel(const int* __restrict__ dst, int E, int* __restrict__ deg) {
    int e = blockIdx.x * blockDim.x + threadIdx.x;
    if (e < E) atomicAdd(&deg[dst[e]], 1);
}

__global__ void dinv_kernel(const int* __restrict__ deg, float* __restrict__ dinv, int N) {
    int i = blockIdx.x * blockDim.x + threadIdx.x;
    if (i < N) {
        float d = (float)(deg[i] + 1);   // +1 self loop; always >= 1
        dinv[i] = 1.0f / sqrtf(d);
    }
}

// out[cols][rows] = in[rows][cols]
__global__ void transpose_kernel(const float* __restrict__ in, float* __restrict__ out,
                                 int rows, int cols) {
    int t = blockIdx.x * blockDim.x + threadIdx.x;
    if (t < rows * cols) {
        int i = t / cols, j = t % cols;
        out[j * rows + i] = in[t];
    }
}

// ---------------- fp32 WMMA GEMM ----------------
// One wave computes a 16x16 output tile of  out = A[nrows x K] * B[K x ncols]
// Bt is B transposed: Bt[ncols][K]  (contiguous along K for v2f fragment loads).
// mode 0: out = lrelu(gemm + bias[col])
// mode 1: out = gemm (raw t) ; out_acc = gemm * dinv[row]^2   (GCN self-loop init)
__global__ __launch_bounds__(256)
void wmma_gemm_kernel(const float* __restrict__ A, const float* __restrict__ Bt,
                      const float* __restrict__ bias, float* __restrict__ out,
                      const float* __restrict__ dinv, float* __restrict__ out_acc,
                      int nrows, int K, int ncols, int mode) {
    int wave = threadIdx.x >> 5;
    int gw   = blockIdx.x * (blockDim.x >> 5) + wave;
    int ct   = ncols >> 4;            // column tiles
    int nrt  = nrows >> 4;            // row tiles (nrows % 16 == 0 here)
    int rt   = gw / ct;
    int c    = gw % ct;
    if (rt >= nrt) return;            // wave-uniform; EXEC stays all-ones past here

    int lane = threadIdx.x & 31;
    int ln   = lane & 15;             // m for A-frag, n for B-frag
    int kb   = (lane >> 4) << 1;      // 0 or 2 per ISA 16x4 f32 layout

    const float* ap = A  + (size_t)(rt * 16 + ln) * K + kb;
    const float* bp = Bt + (size_t)(c  * 16 + ln) * K + kb;

    v8f acc = {};
    #pragma unroll 4
    for (int k = 0; k < K; k += 4) {
        v2f a = *(const v2f*)(ap + k);
        v2f b = *(const v2f*)(bp + k);
        acc = __builtin_amdgcn_wmma_f32_16x16x4_f32(false, a, false, b,
                                                    (short)0, acc, false, false);
    }

    // C/D layout: VGPR r -> row (r) lanes 0-15, row (r+8) lanes 16-31; col = lane&15
    int mbase = rt * 16 + ((lane >> 4) << 3);
    int col   = c * 16 + ln;

    if (mode == 0) {
        float bv = bias[col];
        #pragma unroll
        for (int r = 0; r < 8; ++r) {
            float v = acc[r] + bv;
            out[(size_t)(mbase + r) * ncols + col] = lrelu(v);
        }
    } else {
        #pragma unroll
        for (int r = 0; r < 8; ++r) {
            int   row = mbase + r;
            float t   = acc[r];
            float di  = dinv[row];
            out[(size_t)row * ncols + col]     = t;
            out_acc[(size_t)row * ncols + col] = t * di * di;   // self-loop term
        }
    }
}

// ---------------- GCN edge scatter ----------------
// 16 lanes per edge, float4 per lane: acc[dst] += t[src] * dinv[src]*dinv[dst]
__global__ void edge_scatter_kernel(const int* __restrict__ src, const int* __restrict__ dst,
                                    const float* __restrict__ dinv,
                                    const float* __restrict__ t, float* __restrict__ acc,
                                    int E) {
    int idx = blockIdx.x * blockDim.x + threadIdx.x;
    int e = idx >> 4;
    if (e >= E) return;
    int c = idx & 15;
    int s = src[e], d = dst[e];
    float norm = dinv[s] * dinv[d];
    float4 v = *(const float4*)(t + (size_t)s * D_OUT + c * 4);
    float* o = acc + (size_t)d * D_OUT + c * 4;
    atomicAdd(o + 0, v.x * norm);
    atomicAdd(o + 1, v.y * norm);
    atomicAdd(o + 2, v.z * norm);
    atomicAdd(o + 3, v.w * norm);
}

// h = lrelu(acc + bias)  over N x 64
__global__ void bias_lrelu_kernel(const float* __restrict__ acc, const float* __restrict__ bias,
                                  float* __restrict__ h, int total) {
    int i = blockIdx.x * blockDim.x + threadIdx.x;
    if (i < total) h[i] = lrelu(acc[i] + bias[i & (D_OUT - 1)]);
}

// logits = h @ Wc + bc  (Wc: [64][2]); also copies h into hout (tuple output #2)
__global__ void classifier_kernel(const float* __restrict__ h, const float* __restrict__ Wc,
                                  const float* __restrict__ bc,
                                  float* __restrict__ logits, float* __restrict__ hout, int N) {
    int i = blockIdx.x * blockDim.x + threadIdx.x;
    if (i >= N) return;
    const float4* hr = (const float4*)(h + (size_t)i * D_OUT);
    float4* ho = (float4*)(hout + (size_t)i * D_OUT);
    float l0 = bc[0], l1 = bc[1];
    #pragma unroll
    for (int q = 0; q < 16; ++q) {
        float4 v = hr[q];
        ho[q] = v;
        int k = q * 4;
        l0 += v.x * Wc[2*k+0] + v.y * Wc[2*k+2] + v.z * Wc[2*k+4] + v.w * Wc[2*k+6];
        l1 += v.x * Wc[2*k+1] + v.y * Wc[2*k+3] + v.z * Wc[2*k+5] + v.w * Wc[2*k+7];
    }
    logits[2*i + 0] = l0;
    logits[2*i + 1] = l1;
}

// ---------------- launch ----------------

static inline size_t align256(size_t x) { return (x + 255) & ~(size_t)255; }

extern "C" void kernel_launch(void* const* d_in, const int* in_sizes, int n_in,
                              void* d_out, int out_size, void* d_ws, size_t ws_size,
                              hipStream_t stream) {
    const float* x   = (const float*)d_in[0];
    const int*   ei  = (const int*)  d_in[1];
    const float* W1  = (const float*)d_in[2];
    const float* b1  = (const float*)d_in[3];
    const float* W2  = (const float*)d_in[4];
    const float* b2  = (const float*)d_in[5];
    const float* Wg1 = (const float*)d_in[6];
    const float* bg1 = (const float*)d_in[7];
    const float* Wg2 = (const float*)d_in[8];
    const float* bg2 = (const float*)d_in[9];
    const float* Wc  = (const float*)d_in[10];
    const float* bc  = (const float*)d_in[11];

    const int N = in_sizes[0] / D_IN;     // 100000
    const int E = in_sizes[1] / 2;        // 3200000
    const int* src = ei;
    const int* dstv = ei + E;

    // workspace carve-out
    char* ws = (char*)d_ws;
    size_t off = 0;
    int*   deg  = (int*)  (ws + off); off = align256(off + (size_t)N * 4);
    float* dinv = (float*)(ws + off); off = align256(off + (size_t)N * 4);
    float* W1t  = (float*)(ws + off); off = align256(off + (size_t)D_IN * D_H * 4);
    float* W2t  = (float*)(ws + off); off = align256(off + (size_t)D_H * D_OUT * 4);
    float* Wg1t = (float*)(ws + off); off = align256(off + (size_t)D_OUT * D_OUT * 4);
    float* Wg2t = (float*)(ws + off); off = align256(off + (size_t)D_OUT * D_OUT * 4);
    float* h1   = (float*)(ws + off); off = align256(off + (size_t)N * D_H * 4);
    float* hbuf = (float*)(ws + off); off = align256(off + (size_t)N * D_OUT * 4);
    float* tbuf = (float*)(ws + off); off = align256(off + (size_t)N * D_OUT * 4);
    float* accb = (float*)(ws + off); off = align256(off + (size_t)N * D_OUT * 4);

    float* logits = (float*)d_out;
    float* hout   = (float*)d_out + (size_t)2 * N;

    const int TB = 256;

    // degrees + norm
    fill_zero_u32<<<(N + TB - 1) / TB, TB, 0, stream>>>((unsigned*)deg, N);
    deg_count_kernel<<<(E + TB - 1) / TB, TB, 0, stream>>>(dstv, E, deg);
    dinv_kernel<<<(N + TB - 1) / TB, TB, 0, stream>>>(deg, dinv, N);

    // weight transposes (B^T for contiguous-K WMMA fragments)
    transpose_kernel<<<(D_IN * D_H + TB - 1) / TB, TB, 0, stream>>>(W1, W1t, D_IN, D_H);
    transpose_kernel<<<(D_H * D_OUT + TB - 1) / TB, TB, 0, stream>>>(W2, W2t, D_H, D_OUT);
    transpose_kernel<<<(D_OUT * D_OUT + TB - 1) / TB, TB, 0, stream>>>(Wg1, Wg1t, D_OUT, D_OUT);
    transpose_kernel<<<(D_OUT * D_OUT + TB - 1) / TB, TB, 0, stream>>>(Wg2, Wg2t, D_OUT, D_OUT);

    const int nrt = N / 16;  // 6250 row tiles
    // encoder layer 1: h1 = lrelu(x @ W1 + b1)     [N,512]x[512,128]
    {
        int waves = nrt * (D_H / 16);
        wmma_gemm_kernel<<<(waves + 7) / 8, TB, 0, stream>>>(
            x, W1t, b1, h1, nullptr, nullptr, N, D_IN, D_H, 0);
    }
    // encoder layer 2: hbuf = lrelu(h1 @ W2 + b2)  [N,128]x[128,64]
    {
        int waves = nrt * (D_OUT / 16);
        wmma_gemm_kernel<<<(waves + 7) / 8, TB, 0, stream>>>(
            h1, W2t, b2, hbuf, nullptr, nullptr, N, D_H, D_OUT, 0);
    }

    // ---- GCN layer 1 ----
    {
        int waves = nrt * (D_OUT / 16);
        wmma_gemm_kernel<<<(waves + 7) / 8, TB, 0, stream>>>(
            hbuf, Wg1t, nullptr, tbuf, dinv, accb, N, D_OUT, D_OUT, 1);
        long long st = (long long)E * 16;
        edge_scatter_kernel<<<(int)((st + TB - 1) / TB), TB, 0, stream>>>(
            src, dstv, dinv, tbuf, accb, E);
        bias_lrelu_kernel<<<(N * D_OUT + TB - 1) / TB, TB, 0, stream>>>(
            accb, bg1, hbuf, N * D_OUT);
    }
    // ---- GCN layer 2 ----
    {
        int waves = nrt * (D_OUT / 16);
        wmma_gemm_kernel<<<(waves + 7) / 8, TB, 0, stream>>>(
            hbuf, Wg2t, nullptr, tbuf, dinv, accb, N, D_OUT, D_OUT, 1);
        long long st = (long long)E * 16;
        edge_scatter_kernel<<<(int)((st + TB - 1) / TB), TB, 0, stream>>>(
            src, dstv, dinv, tbuf, accb, E);
        bias_lrelu_kernel<<<(N * D_OUT + TB - 1) / TB, TB, 0, stream>>>(
            accb, bg2, hbuf, N * D_OUT);
    }

    // classifier + h copy (tuple output: logits then h)
    classifier_kernel<<<(N + TB - 1) / TB, TB, 0, stream>>>(hbuf, Wc, bc, logits, hout, N);
}